// SparseGCN_2121713845070
// MI455X (gfx1250) — compile-verified
//
#include <hip/hip_runtime.h>

#define N_NODES 50000
#define N_EDGES 800000
#define IN_DIM 128
#define HIDDEN 128
#define OUT_DIM 40

typedef __attribute__((ext_vector_type(2))) float v2f;
typedef __attribute__((ext_vector_type(8))) float v8f;

// ---------------------------------------------------------------------------
// Zero a buffer (needed before float-atomic accumulation; ws is poisoned).
// ---------------------------------------------------------------------------
__global__ __launch_bounds__(256) void zero_f4_kernel(float4* __restrict__ p, int n4) {
    int i = blockIdx.x * blockDim.x + threadIdx.x;
    int stride = gridDim.x * blockDim.x;
    for (; i < n4; i += stride) {
        p[i] = make_float4(0.f, 0.f, 0.f, 0.f);
    }
}

// ---------------------------------------------------------------------------
// SPMM: out[dst[e], :] += w[e] * x[src[e], :]   (D = 128)
// One wave32 per edge; each lane owns 4 consecutive features (float4 gather,
// 4x global_atomic_add_f32 scatter). L2 (192MB) holds the 25.6MB feature
// matrix, so random gathers/atomics are L2-resident.
// ---------------------------------------------------------------------------
__global__ __launch_bounds__(256) void spmm_kernel(const int* __restrict__ esrc,
                                                   const int* __restrict__ edst,
                                                   const float* __restrict__ ew,
                                                   const float* __restrict__ x,
                                                   float* __restrict__ out) {
    const int e = blockIdx.x * 8 + (threadIdx.x >> 5);
    if (e >= N_EDGES) return;
    const int lane = threadIdx.x & 31;

    const int s = esrc[e];
    const int d = edst[e];
    const float w = ew[e];

    const float4 v = *(const float4*)(x + (size_t)s * HIDDEN + lane * 4);
    float* o = out + (size_t)d * HIDDEN + lane * 4;

    // Hardware FP32 atomics (global_atomic_add_f32), no CAS loop.
    unsafeAtomicAdd(o + 0, w * v.x);
    unsafeAtomicAdd(o + 1, w * v.y);
    unsafeAtomicAdd(o + 2, w * v.z);
    unsafeAtomicAdd(o + 3, w * v.w);
}

// ---------------------------------------------------------------------------
// Dense GEMM via V_WMMA_F32_16X16X4_F32:  C[M,NCOLS] = A[M,128] * B[128,NCOLS]
// + bias (+ optional ReLU).  One wave computes one 16x16 output tile,
// iterating K=128 in steps of 4 (32 WMMA issues per tile).
//
// Fragment layouts per CDNA5 ISA 7.12.2 (f32, wave32):
//   A 16x4 : lane -> M = lane&15; vgpr pair holds (K+2*half, K+2*half+1)
//   B 4x16 : lane -> N = lane&15; vgpr pair holds same K pair
//   C/D    : vgpr r -> M = r + 8*half, N = lane&15
// ---------------------------------------------------------------------------
template <int NWAVES, int NCOLS, bool RELU>
__global__ __launch_bounds__(NWAVES * 32) void gemm_wmma_kernel(
        const float* __restrict__ A,     // [M, 128] row-major
        const float* __restrict__ B,     // [128, NCOLS] row-major
        const float* __restrict__ bias,  // [NCOLS]
        float* __restrict__ C) {         // [M, NCOLS] row-major
    const int wave = threadIdx.x >> 5;
    const int lane = threadIdx.x & 31;
    const int half = lane >> 4;   // 0: lanes 0-15, 1: lanes 16-31
    const int l16  = lane & 15;

    const int m0 = blockIdx.x * 16;   // M tile (50000 % 16 == 0: no guard)
    const int n0 = wave * 16;         // N tile

    const int ncol  = n0 + l16;
    const int ncolc = (ncol < NCOLS) ? ncol : (NCOLS - 1);  // clamp load col;
    // clamped lanes feed only output columns >= NCOLS which are never stored
    // (WMMA mixes across K, not across N).

    const float* arow = A + (size_t)(m0 + l16) * 128 + 2 * half;

    v8f c = {};
#pragma unroll 4
    for (int k0 = 0; k0 < 128; k0 += 4) {
        // A fragment: two consecutive K values -> one aligned float2 load.
        v2f a = *(const v2f*)(arow + k0);
        // B fragment: same K pair at column ncolc (stride NCOLS between rows).
        const float* bp = B + (size_t)(k0 + 2 * half) * NCOLS + ncolc;
        v2f b;
        b.x = bp[0];
        b.y = bp[NCOLS];
        c = __builtin_amdgcn_wmma_f32_16x16x4_f32(
                /*neg_a=*/false, a, /*neg_b=*/false, b,
                /*c_mod=*/(short)0, c, /*reuse_a=*/false, /*reuse_b=*/false);
    }

    const float bb = (ncol < NCOLS) ? bias[ncol] : 0.f;
#pragma unroll
    for (int r = 0; r < 8; ++r) {
        const int m = m0 + r + half * 8;
        if (ncol < NCOLS) {
            float v = c[r] + bb;
            if (RELU) v = fmaxf(v, 0.f);
            C[(size_t)m * NCOLS + ncol] = v;
        }
    }
}

// ---------------------------------------------------------------------------
// Launch: zero(A); spmm(x->A); gemm1(A->B, relu); zero(A); spmm(B->A);
//         gemm2(A->out).
// ---------------------------------------------------------------------------
extern "C" void kernel_launch(void* const* d_in, const int* in_sizes, int n_in,
                              void* d_out, int out_size, void* d_ws, size_t ws_size,
                              hipStream_t stream) {
    (void)in_sizes; (void)n_in; (void)out_size; (void)ws_size;

    const float* x    = (const float*)d_in[0];
    const int*   esrc = (const int*)d_in[1];
    const int*   edst = (const int*)d_in[2];
    const float* ew   = (const float*)d_in[3];
    const float* W1   = (const float*)d_in[4];
    const float* b1   = (const float*)d_in[5];
    const float* W2   = (const float*)d_in[6];
    const float* b2   = (const float*)d_in[7];
    float* out = (float*)d_out;

    float* bufA = (float*)d_ws;                      // 50000*128 f32 = 25.6 MB
    float* bufB = bufA + (size_t)N_NODES * HIDDEN;   // 25.6 MB

    const int n4 = N_NODES * HIDDEN / 4;
    const int spmm_blocks = (N_EDGES + 7) / 8;
    const int mtiles = N_NODES / 16;  // 3125, exact

    zero_f4_kernel<<<2048, 256, 0, stream>>>((float4*)bufA, n4);
    spmm_kernel<<<spmm_blocks, 256, 0, stream>>>(esrc, edst, ew, x, bufA);
    gemm_wmma_kernel<8, HIDDEN, true><<<mtiles, 256, 0, stream>>>(bufA, W1, b1, bufB);
    zero_f4_kernel<<<2048, 256, 0, stream>>>((float4*)bufA, n4);
    spmm_kernel<<<spmm_blocks, 256, 0, stream>>>(esrc, edst, ew, bufB, bufA);
    gemm_wmma_kernel<3, OUT_DIM, false><<<mtiles, 96, 0, stream>>>(bufA, W2, b2, out);
}